// LuGreCell_80152679678111
// MI455X (gfx1250) — compile-verified
//
#include <hip/hip_runtime.h>
#include <math.h>

typedef float v2f __attribute__((ext_vector_type(2)));
typedef float v8f __attribute__((ext_vector_type(8)));

// gfx1250 hardware transcendentals (v_tanh_f32 / v_exp_f32 / v_log_f32 / v_rcp_f32)
#if __has_builtin(__builtin_amdgcn_tanhf)
#define TANHF(x) __builtin_amdgcn_tanhf(x)
#else
#define TANHF(x) tanhf(x)
#endif
#if __has_builtin(__builtin_amdgcn_exp2f)
#define EXP2F(x) __builtin_amdgcn_exp2f(x)
#else
#define EXP2F(x) exp2f(x)
#endif
#if __has_builtin(__builtin_amdgcn_logf)
#define LOG2F(x) __builtin_amdgcn_logf(x)
#else
#define LOG2F(x) log2f(x)
#endif
#if __has_builtin(__builtin_amdgcn_rcpf)
#define RCPF(x) __builtin_amdgcn_rcpf(x)
#else
#define RCPF(x) (1.0f / (x))
#endif

#define LOG2E 1.44269504088896340736f
#define LN2   0.69314718055994530942f
#define DT_K  1e-3f

// One wave32 owns 16 batch streams for all T steps.
// GEMM1 (mi@W1 + b1) = 4x V_WMMA_F32_16X16X4_F32 with A=W1^T (+bias as K=2),
// B built lane-locally from (v, sz). GEMM2 = per-lane dot (8 parallel
// accumulator chains -> tree reduce) + one shfl_xor(16) cross-half add.
__global__ __launch_bounds__(256, 1) void lugre_wmma_kernel(
    const float* __restrict__ x,
    const float* __restrict__ p_s1, const float* __restrict__ p_s2,
    const float* __restrict__ p_alpha, const float* __restrict__ p_vs,
    const float* __restrict__ W1, const float* __restrict__ b1,
    const float* __restrict__ W2, const float* __restrict__ b2,
    float* __restrict__ out, int B, int T)
{
    const int lane = threadIdx.x & 31;
    const int wave = blockIdx.x * (blockDim.x >> 5) + (threadIdx.x >> 5);
    if (wave * 16 >= B) return;            // wave-uniform guard (B multiple of 16)

    const int  col  = lane & 15;           // batch column within the tile
    const int  bidx = wave * 16 + col;
    const bool lo   = lane < 16;
    const int  hsel = (lane & 16) >> 1;    // 0 for lanes 0-15, 8 for lanes 16-31

    const float s1    = fabsf(p_s1[0]);
    const float s2    = fabsf(p_s2[0]);
    const float alpha = p_alpha[0];
    const float v_s   = p_vs[0];
    const float bb2   = b2[0];

    // A-matrix tiles (loop invariant). A layout (f32 16x4): lane m = row m.
    // lanes 0-15:  V0 = K0 = W1[0][h] (v coeff),  V1 = K1 = W1[1][h] (sz coeff)
    // lanes 16-31: V0 = K2 = b1[h]   (bias),      V1 = K3 = 0
    v2f  wa[4];
    float w2v[32];   // per-lane slice of W2 matching the C-tile layout
#pragma unroll
    for (int r = 0; r < 4; ++r) {
        int hid = r * 16 + col;
        wa[r][0] = lo ? W1[hid]      : b1[hid];
        wa[r][1] = lo ? W1[64 + hid] : 0.0f;
#pragma unroll
        for (int j = 0; j < 8; ++j)
            w2v[r * 8 + j] = W2[r * 16 + hsel + j];
    }

    const float* xb = x + (size_t)bidx * T * 3;
    float cv = xb[0], cfc = xb[1], cfs = xb[2];   // prefetched t=0 triple
    float sz = 0.0f;                               // recurrent state (mirrored in both halves)

    for (int t = 0; t < T; ++t) {
        // ---- prefetch next step's (v, F_c, F_s) ----
        int tn = (t + 1 < T) ? (t + 1) : t;        // wave-uniform clamp (SALU)
        float nv  = xb[3 * tn];
        float nfc = xb[3 * tn + 1];
        float nfs = xb[3 * tn + 2];

        // ---- GEMM1: H_pre = W1^T . [v; sz; 1; 0]  (4 WMMA f32 tiles) ----
        // B layout (f32 4x16): lane n = column n.
        // lanes 0-15: V0=K0=v, V1=K1=sz ; lanes 16-31: V0=K2=1 (bias), V1=K3=0
        v2f bm;
        bm[0] = lo ? cv : 1.0f;
        bm[1] = lo ? sz : 0.0f;
        v8f cz = {};
        v8f d0 = __builtin_amdgcn_wmma_f32_16x16x4_f32(false, wa[0], false, bm, (short)0, cz, false, false);
        v8f d1 = __builtin_amdgcn_wmma_f32_16x16x4_f32(false, wa[1], false, bm, (short)0, cz, false, false);
        v8f d2 = __builtin_amdgcn_wmma_f32_16x16x4_f32(false, wa[2], false, bm, (short)0, cz, false, false);
        v8f d3 = __builtin_amdgcn_wmma_f32_16x16x4_f32(false, wa[3], false, bm, (short)0, cz, false, false);

        // ---- tanh: 32 independent temporaries (lets trans ops overlap fmas) ----
        float h[32];
#pragma unroll
        for (int j = 0; j < 8; ++j) h[j]      = TANHF(d0[j]);
#pragma unroll
        for (int j = 0; j < 8; ++j) h[8 + j]  = TANHF(d1[j]);
#pragma unroll
        for (int j = 0; j < 8; ++j) h[16 + j] = TANHF(d2[j]);
#pragma unroll
        for (int j = 0; j < 8; ++j) h[24 + j] = TANHF(d3[j]);

        // ---- GEMM2 dot with W2: 8 parallel accumulator chains ----
        float ac[8];
#pragma unroll
        for (int j = 0; j < 8; ++j) ac[j] = h[j] * w2v[j];
#pragma unroll
        for (int r = 1; r < 4; ++r)
#pragma unroll
            for (int j = 0; j < 8; ++j)
                ac[j] = fmaf(h[r * 8 + j], w2v[r * 8 + j], ac[j]);
        float sA  = (ac[0] + ac[1]) + (ac[2] + ac[3]);
        float sB  = (ac[4] + ac[5]) + (ac[6] + ac[7]);
        float acc = sA + sB;
        float raw = acc + __shfl_xor(acc, 16, 32) + bb2;   // both halves get full sum

        // ---- softplus(raw) ----
        float et  = EXP2F(-fabsf(raw) * LOG2E);
        float sp  = fmaxf(raw, 0.0f) + LOG2F(1.0f + et) * LN2;

        // ---- LuGre elementwise update (fp32, mirrored in both halves) ----
        float av    = fabsf(cv);
        float ratio = av * RCPF(v_s);
        float p     = EXP2F(alpha * LOG2F(ratio));        // |v/v_s|^alpha (v->0 => 0)
        float eg    = EXP2F(-p * LOG2E);                  // exp(-p)
        float g     = fmaf(cfs - cfc, eg, cfc);           // F_c + (F_s-F_c)*exp(-|v/vs|^a)
        float rg    = RCPF(g);
        float sgn   = (cv > 0.0f) ? 1.0f : ((cv < 0.0f) ? -1.0f : 0.0f);
        float k     = g * sgn;
        float E     = EXP2F(-DT_K * LOG2E * sp * av * rg);
        float szn   = fmaf(sz - k, E, k);
        float zdot  = fmaf(-szn * av, rg, cv);            // uses pre-clip sz
        float szc   = fminf(fmaxf(szn, -cfs), cfs);
        float F     = fmaf(s1, zdot, fmaf(s2, cv, szc));
        sz = szc;

        if (lo) out[(size_t)bidx * T + t] = F;

        cv = nv; cfc = nfc; cfs = nfs;
    }
}

extern "C" void kernel_launch(void* const* d_in, const int* in_sizes, int n_in,
                              void* d_out, int out_size, void* d_ws, size_t ws_size,
                              hipStream_t stream) {
    (void)n_in; (void)d_ws; (void)ws_size;
    const float* x   = (const float*)d_in[0];
    const float* s1  = (const float*)d_in[1];
    const float* s2  = (const float*)d_in[2];
    const float* al  = (const float*)d_in[3];
    const float* vs  = (const float*)d_in[4];
    const float* W1  = (const float*)d_in[5];
    const float* b1  = (const float*)d_in[6];
    const float* W2  = (const float*)d_in[7];
    const float* b2  = (const float*)d_in[8];
    float* out = (float*)d_out;

    const int T = 2048;                      // reference: x is [B, T, 3]
    const int B = in_sizes[0] / (3 * T);     // 4096
    (void)out_size;

    dim3 block(256);                          // 8 waves/block, 128 batch/block
    dim3 grid((B + 127) / 128);
    hipLaunchKernelGGL(lugre_wmma_kernel, grid, block, 0, stream,
                       x, s1, s2, al, vs, W1, b1, W2, b2, out, B, T);
}